// DGLGat_45569603011006
// MI455X (gfx1250) — compile-verified
//
#include <hip/hip_runtime.h>
#include <cstdint>
#include <cstddef>

#define GAT_N_NODES 50000
#define GAT_N_EDGES 800000
#define NEG_SLOPE 0.2f
#define K_MAX 256
#define LDS_PAD 2   // pad LDS row stride (words) to spread rows across banks

typedef float v2f __attribute__((ext_vector_type(2)));
typedef float v8f __attribute__((ext_vector_type(8)));

#if defined(__has_builtin)
#if __has_builtin(__builtin_amdgcn_global_load_async_to_lds_b128)
#define HAVE_ASYNC_LDS 1
// per hipcc diagnostic: param0 is "int __vector_size__(16) __device__ *"
typedef int v4i_vs __attribute__((vector_size(16)));
typedef v4i_vs __attribute__((address_space(1))) gv4i;   // global
typedef v4i_vs __attribute__((address_space(3))) lv4i;   // LDS
#endif
#endif

// ---------------------------------------------------------------------------
// GEMM: H[M,Nf] = X[M,K] @ W[K,Nf] with V_WMMA_F32_16X16X4_F32 (full f32).
// One block per 16-row M tile; one wave per 16-col N tile (blockDim = ntiles*32).
// The 16xK X tile is staged once into LDS (async-to-LDS when available) and
// shared by all N-tile waves; A fragments then come from ds_load_b64.
// ISA A-layout (16x4 f32): lanes 0-15 hold K=0 (v0), K=1 (v1); lanes 16-31
// hold K=2 (v0), K=3 (v1). B mirrored. C/D: VGPR r -> row r / r+8.
// ---------------------------------------------------------------------------
__global__ void gat_gemm_wmma(const float* __restrict__ X,
                              const float* __restrict__ W,
                              float* __restrict__ H,
                              int M, int K, int Nf)
{
    __shared__ float Xs[16 * (K_MAX + LDS_PAD)];
    const int stride = K + LDS_PAD;
    const int m0  = blockIdx.x << 4;
    const int tid = threadIdx.x;
    const int nthreads = blockDim.x;

    // ---- cooperative stage of X[m0:m0+16, 0:K] into LDS (16B per thread) ----
    const int segs_per_row = K >> 2;           // float4 segments per row
    const int nseg = 16 * segs_per_row;        // multiple of nthreads here
    for (int s = tid; s < nseg; s += nthreads) {
        const int r = s / segs_per_row;
        const int c = (s - r * segs_per_row) << 2;
        const float* gsrc = X + (size_t)(m0 + r) * K + c;
        float* lds = &Xs[r * stride + c];
#if defined(HAVE_ASYNC_LDS)
        __builtin_amdgcn_global_load_async_to_lds_b128((gv4i*)gsrc, (lv4i*)lds, 0, 0);
#else
        const float4 v = *(const float4*)gsrc;
        lds[0] = v.x; lds[1] = v.y; lds[2] = v.z; lds[3] = v.w;
#endif
    }
#if defined(HAVE_ASYNC_LDS)
#if __has_builtin(__builtin_amdgcn_s_wait_asynccnt)
    __builtin_amdgcn_s_wait_asynccnt(0);
#else
    asm volatile("s_wait_asynccnt 0" ::: "memory");
#endif
#endif
    __syncthreads();

    // ---- per-wave 16x16 WMMA tile ----
    const int waveid = tid >> 5;               // N tile index
    const int lane   = tid & 31;
    const int n0  = waveid << 4;
    const int row = lane & 15;                 // M row (A) / N col (B,C)
    const int khalf = (lane >> 4) << 1;        // 0 (lanes 0-15) or 2 (lanes 16-31)

    const float* Ap  = &Xs[row * stride + khalf];
    const float* Wp0 = W + (size_t)khalf * Nf + n0 + row;       // row k+khalf
    const float* Wp1 = Wp0 + Nf;                                // row k+khalf+1
    const size_t wstep = (size_t)4 * Nf;

    v8f acc = {0.f, 0.f, 0.f, 0.f, 0.f, 0.f, 0.f, 0.f};
    #pragma unroll 4
    for (int k = 0; k < K; k += 4) {
        v2f a, b;
        a.x = Ap[k];                           // ds_load_b64 from staged tile
        a.y = Ap[k + 1];
        b.x = *Wp0;
        b.y = *Wp1;
        Wp0 += wstep;
        Wp1 += wstep;
        acc = __builtin_amdgcn_wmma_f32_16x16x4_f32(false, a, false, b,
                                                    (short)0, acc, false, false);
    }

    const int rbase = m0 + ((lane >> 4) << 3); // +8 rows for upper half-wave
    float* Hp = H + (size_t)rbase * Nf + n0 + row;
    #pragma unroll
    for (int r = 0; r < 8; ++r)
        Hp[(size_t)r * Nf] = acc[r];
}

// el[n,h] = <h[n,h,:], al[h,:]> ; er likewise
__global__ void gat_eler(const float* __restrict__ H,
                         const float* __restrict__ al,
                         const float* __restrict__ ar,
                         float* __restrict__ el, float* __restrict__ er,
                         int n_nodes, int heads, int D)
{
    int i = blockIdx.x * blockDim.x + threadIdx.x;
    if (i >= n_nodes * heads) return;
    int n = i / heads, h = i - n * heads;
    const float* hp  = H + ((size_t)n * heads + h) * D;
    const float* alp = al + (size_t)h * D;
    const float* arp = ar + (size_t)h * D;
    float sl = 0.f, sr = 0.f;
    for (int d = 0; d < D; ++d) {
        float v = hp[d];
        sl += v * alp[d];
        sr += v * arp[d];
    }
    el[i] = sl;
    er[i] = sr;
}

// order-preserving float<->uint key for atomic max (init key = 0 < key(-inf))
__device__ __forceinline__ unsigned f2key(float f) {
    unsigned u = __float_as_uint(f);
    return (u >> 31) ? ~u : (u | 0x80000000u);
}
__device__ __forceinline__ float key2f(unsigned k) {
    unsigned u = (k & 0x80000000u) ? (k & 0x7FFFFFFFu) : ~k;
    return __uint_as_float(u);
}

// pass 1: e = leaky_relu(el[src]+er[dst]); segment max over dst via atomicMax
__global__ void gat_edge1(const int* __restrict__ src, const int* __restrict__ dst,
                          const float* __restrict__ el, const float* __restrict__ er,
                          float* __restrict__ eWS, unsigned* __restrict__ mkey,
                          int n_edges, int heads)
{
    int i = blockIdx.x * blockDim.x + threadIdx.x;
    if (i >= n_edges * heads) return;
    int e = i / heads, h = i - e * heads;
    int s = src[e], d = dst[e];
    float v = el[s * heads + h] + er[d * heads + h];
    v = (v > 0.f) ? v : NEG_SLOPE * v;
    eWS[i] = v;
    atomicMax(&mkey[d * heads + h], f2key(v));
}

// pass 2: ex = exp(e - m[dst]); segment sum -> denom
__global__ void gat_edge2(const int* __restrict__ dst,
                          const unsigned* __restrict__ mkey,
                          float* __restrict__ eWS, float* __restrict__ denom,
                          int n_edges, int heads)
{
    int i = blockIdx.x * blockDim.x + threadIdx.x;
    if (i >= n_edges * heads) return;
    int e = i / heads, h = i - e * heads;
    int d = dst[e];
    float m  = key2f(mkey[d * heads + h]);
    float ex = __expf(eWS[i] - m);
    eWS[i] = ex;
    atomicAdd(&denom[d * heads + h], ex);
}

// pass 3: out[dst] += (ex/denom[dst]) * h[src]; one thread per (edge,head,feat)
__global__ void gat_edge3(const int* __restrict__ src, const int* __restrict__ dst,
                          const float* __restrict__ eWS, const float* __restrict__ denom,
                          const float* __restrict__ H, float* __restrict__ out,
                          int n_edges, int heads, int D)
{
    long long i = (long long)blockIdx.x * blockDim.x + threadIdx.x;
    long long total = (long long)n_edges * heads * D;
    if (i >= total) return;
    int dd = (int)(i % D);
    long long eh = i / D;
    int h = (int)(eh % heads);
    int e = (int)(eh / heads);
    int s = src[e], d = dst[e];
    float alpha = eWS[e * heads + h] / denom[d * heads + h];
    float msg = alpha * H[((size_t)s * heads + h) * D + dd];
    atomicAdd(&out[((size_t)d * heads + h) * D + dd], msg);
}

// out = (relu?) (agg + bias)   (in-place safe)
__global__ void gat_bias_act(const float* __restrict__ agg,
                             const float* __restrict__ bias,
                             float* __restrict__ out,
                             int n_nodes, int F, int do_relu)
{
    int i = blockIdx.x * blockDim.x + threadIdx.x;
    if (i >= n_nodes * F) return;
    int f = i % F;
    float v = agg[i] + bias[f];
    if (do_relu) v = (v > 0.f) ? v : 0.f;
    out[i] = v;
}

// ---------------------------------------------------------------------------
extern "C" void kernel_launch(void* const* d_in, const int* in_sizes, int n_in,
                              void* d_out, int out_size, void* d_ws, size_t ws_size,
                              hipStream_t stream)
{
    (void)in_sizes; (void)n_in; (void)out_size; (void)ws_size;

    const float* feat = (const float*)d_in[0];
    const int*   src  = (const int*)d_in[1];
    const int*   dst  = (const int*)d_in[2];
    const float* W1 = (const float*)d_in[3];
    const float* al1 = (const float*)d_in[4];
    const float* ar1 = (const float*)d_in[5];
    const float* b1 = (const float*)d_in[6];
    const float* W2 = (const float*)d_in[7];
    const float* al2 = (const float*)d_in[8];
    const float* ar2 = (const float*)d_in[9];
    const float* b2 = (const float*)d_in[10];
    const float* W3 = (const float*)d_in[11];
    const float* al3 = (const float*)d_in[12];
    const float* ar3 = (const float*)d_in[13];
    const float* b3 = (const float*)d_in[14];

    // workspace layout (floats)
    float* ws   = (float*)d_ws;
    float* A    = ws;                                   // [N,128] gemm output (h)
    float* B    = A   + (size_t)GAT_N_NODES * 128;      // [N,128] aggregated / next x
    float* eWS  = B   + (size_t)GAT_N_NODES * 128;      // [E,4]   edge logits / exp
    float* el   = eWS + (size_t)GAT_N_EDGES * 4;        // [N,4]
    float* er   = el  + (size_t)GAT_N_NODES * 4;        // [N,4]
    unsigned* mkey = (unsigned*)(er + (size_t)GAT_N_NODES * 4);  // [N,4]
    float* denom   = (float*)(mkey + (size_t)GAT_N_NODES * 4);   // [N,4]

    const int TB = 256;

    auto run_layer = [&](const float* x, int K,
                         const float* W, const float* al, const float* ar,
                         const float* bias, int heads, int D,
                         float* outbuf, int do_relu)
    {
        const int Nf     = heads * D;
        const int ntiles = Nf / 16;            // waves per block (8 or 4)
        const int mblk   = GAT_N_NODES / 16;   // 3125 blocks, exact
        gat_gemm_wmma<<<mblk, 32 * ntiles, 0, stream>>>(x, W, A, GAT_N_NODES, K, Nf);

        const int nh = GAT_N_NODES * heads;
        gat_eler<<<(nh + TB - 1) / TB, TB, 0, stream>>>(A, al, ar, el, er,
                                                        GAT_N_NODES, heads, D);

        (void)hipMemsetAsync(mkey,  0, (size_t)nh * sizeof(unsigned), stream);
        (void)hipMemsetAsync(denom, 0, (size_t)nh * sizeof(float), stream);
        (void)hipMemsetAsync(outbuf, 0, (size_t)GAT_N_NODES * Nf * sizeof(float), stream);

        const int EH = GAT_N_EDGES * heads;
        gat_edge1<<<(EH + TB - 1) / TB, TB, 0, stream>>>(src, dst, el, er, eWS, mkey,
                                                         GAT_N_EDGES, heads);
        gat_edge2<<<(EH + TB - 1) / TB, TB, 0, stream>>>(dst, mkey, eWS, denom,
                                                         GAT_N_EDGES, heads);
        const long long tot = (long long)GAT_N_EDGES * heads * D;
        gat_edge3<<<(unsigned)((tot + TB - 1) / TB), TB, 0, stream>>>(
            src, dst, eWS, denom, A, outbuf, GAT_N_EDGES, heads, D);

        const int nf = GAT_N_NODES * Nf;
        gat_bias_act<<<(nf + TB - 1) / TB, TB, 0, stream>>>(outbuf, bias, outbuf,
                                                            GAT_N_NODES, Nf, do_relu);
    };

    // layer 1: feat[N,256] -> B[N,128] (4 heads x 32), relu
    run_layer(feat, 256, W1, al1, ar1, b1, 4, 32, B, 1);
    // layer 2: B[N,128] -> B[N,128] (gemm reads B fully before B is re-zeroed)
    run_layer(B, 128, W2, al2, ar2, b2, 4, 32, B, 1);
    // layer 3: B[N,128] -> d_out[N,64] (1 head x 64), no relu
    run_layer(B, 128, W3, al3, ar3, b3, 1, 64, (float*)d_out, 0);
}